// DotProductAttention_1554778161228
// MI455X (gfx1250) — compile-verified
//
#include <hip/hip_runtime.h>

typedef __attribute__((ext_vector_type(2))) float        v2f;
typedef __attribute__((ext_vector_type(8))) float        v8f;
typedef __attribute__((ext_vector_type(4))) unsigned int v4u;
typedef __attribute__((ext_vector_type(8))) int          v8i;
typedef __attribute__((ext_vector_type(4))) int          v4i;

#define B_   16
#define Q_   2048
#define K_   2048
#define D_   128
#define NEGV (-1000000.0f)
#define SCALE 0.08838834764831845f /* 1/sqrt(128) */

// LDS row strides (floats), produced for free by TDM row padding:
//  K: 132 -> b64 B-operand reads hit banks 4m+{0..3}: all 64 banks, no conflict
//  V: 136 -> the two half-waves of a b32 read are 16 banks apart, no conflict
#define KSTR 132
#define VSTR 136

// Issue one TDM 2D tile load: 16 rows x 128 f32 from row-major global into
// LDS with per-row padding. D# packed per CDNA5 ISA.
__device__ __forceinline__ void tdm_load_tile(const float* gsrc, float* ldst,
                                              int pad_amount_code) {
    unsigned long long ga = (unsigned long long)(size_t)gsrc;
    v4u g0;
    g0.x = 1u;                                   // count=1, user descriptor
    g0.y = (unsigned)(size_t)ldst;               // LDS byte offset (low 32 bits)
    g0.z = (unsigned)ga;                         // global_addr[31:0]
    g0.w = (unsigned)((ga >> 32) & 0x01FFFFFFull) | (2u << 30); // [56:32]|type=2
    v8i g1;
    g1[0] = (2 << 16) | (1 << 20) | (6 << 22) | (pad_amount_code << 25);
    g1[1] = (int)(128u << 16);                   // tensor_dim0 = 128 (bits 79:48)
    g1[2] = (int)(2048u << 16);                  // tensor_dim1 = 2048 (bits 111:80)
    g1[3] = (int)(128u << 16);                   // tile_dim0 = 128 (bits 127:112)
    g1[4] = 16;                                  // tile_dim1 = 16 rows
    g1[5] = 128;                                 // tensor_dim0_stride = 128
    g1[6] = 0;
    g1[7] = 0;
    v4i z4 = (v4i)0;                             // 2D tensor: groups 2/3 unused
    v8i z8 = (v8i)0;
    __builtin_amdgcn_tensor_load_to_lds(g0, g1, z4, z4, z8, 0);
}

// One wave owns a 16-row query tile; 8 waves per block share one batch and
// stream shared K/V tiles staged in LDS by the Tensor Data Mover (double
// buffered, TENSORcnt-tracked). All matrix math is V_WMMA_F32_16X16X4_F32,
// with K/V operands software-pipelined through register ping-pong groups so
// ds_load latency is hidden behind the WMMA chains.
__global__ __launch_bounds__(256) void attn_f32_wmma_tdm_kernel(
    const float* __restrict__ qg,
    const float* __restrict__ kg,
    const float* __restrict__ vg,
    const int*   __restrict__ valid_lens,
    float*       __restrict__ outg)
{
    __shared__ float kbuf[2][16 * KSTR];  // 2 x 8448 B
    __shared__ float vbuf[2][16 * VSTR];  // 2 x 8704 B
    __shared__ float pbuf[8][16 * 18];    // per-wave P transpose staging

    const int lane  = threadIdx.x & 31;
    const int wib   = threadIdx.x >> 5;
    const int w     = blockIdx.x * 8 + wib;   // global wave id
    const int b     = w >> 7;                 // one batch per block (8 q-tiles)
    const int qbase = (w & 127) << 4;
    const int m     = lane & 15;
    const int hi    = lane >> 4;

    const float* Qb = qg + ((size_t)b * Q_ + qbase) * D_;
    const float* Kb = kg + (size_t)b * K_ * D_;
    const float* Vb = vg + (size_t)b * K_ * D_;
    float*       Ob = outg + ((size_t)b * Q_ + qbase) * D_;

    const int vlen = valid_lens[b];
    // keys >= vlen contribute exp(-1e6 - m) == 0 exactly when vlen >= 1;
    // vlen == 0 runs every tile (uniform softmax -> mean of V).
    int kmax = (vlen == 0) ? K_ : ((vlen + 15) & ~15);
    if (kmax > K_) kmax = K_;
    const int ntiles = kmax >> 4;             // uniform across the block

    // prologue: TDM-fetch tile 0 (wave 0 issues; TENSORcnt += 2)
    if (wib == 0) {
        tdm_load_tile(Kb, &kbuf[0][0], 3);
        tdm_load_tile(Vb, &vbuf[0][0], 7);
    }

    // Q tile (16x128) straight into WMMA-A layout: a b64 load of two
    // consecutive d values IS the per-lane A vgpr pair.
    v2f qa[32];
#pragma unroll
    for (int kk = 0; kk < 32; ++kk)
        qa[kk] = *(const v2f*)(Qb + (size_t)m * D_ + 4 * kk + 2 * hi);

    v8f   acc[8];
    float mrow[8], lrow[8];
#pragma unroll
    for (int t = 0; t < 8; ++t) acc[t] = (v8f)0.0f;
#pragma unroll
    for (int r = 0; r < 8; ++r) { mrow[r] = -INFINITY; lrow[r] = 0.0f; }

    float* pw = &pbuf[wib][0];

    for (int i = 0; i < ntiles; ++i) {
        const int k0 = i << 4;
        if (wib == 0) {
            if (i + 1 < ntiles) {
                // prefetch tile i+1, then wait so only those 2 remain in
                // flight -> tile i's transfers (in-order per wave) are done
                tdm_load_tile(Kb + (size_t)(k0 + 16) * D_, &kbuf[(i + 1) & 1][0], 3);
                tdm_load_tile(Vb + (size_t)(k0 + 16) * D_, &vbuf[(i + 1) & 1][0], 7);
                __builtin_amdgcn_s_wait_tensorcnt(2);
            } else {
                __builtin_amdgcn_s_wait_tensorcnt(0);
            }
        }
        __syncthreads();                       // publish tile i to all waves
        const float* ks = &kbuf[i & 1][0];
        const float* vs = &vbuf[i & 1][0];

        // ---- S = Q * K^T : two WMMA chains, K operands pipelined in groups
        //      of 8 chunks so ds_loads run a full group ahead of their use ----
        const float* kt = ks + m * KSTR + 2 * hi;
        v2f bkA[8], bkB[8];
        v8f s0 = (v8f)0.0f, s1 = (v8f)0.0f;
#pragma unroll
        for (int j = 0; j < 8; ++j)
            bkA[j] = *(const v2f*)(kt + 4 * j);
#pragma unroll
        for (int g = 0; g < 4; ++g) {
            v2f*       cur = (g & 1) ? bkB : bkA;
            v2f*       nxt = (g & 1) ? bkA : bkB;
            if (g < 3) {
#pragma unroll
                for (int j = 0; j < 8; ++j)
                    nxt[j] = *(const v2f*)(kt + 4 * ((g + 1) * 8 + j));
            }
#pragma unroll
            for (int j = 0; j < 8; j += 2) {
                s0 = __builtin_amdgcn_wmma_f32_16x16x4_f32(
                         false, qa[g * 8 + j], false, cur[j], (short)0, s0,
                         false, false);
                s1 = __builtin_amdgcn_wmma_f32_16x16x4_f32(
                         false, qa[g * 8 + j + 1], false, cur[j + 1], (short)0,
                         s1, false, false);
            }
        }
        v8f s = s0 + s1;

        // ---- scale + key mask (this lane's key column = k0 + m) ----
        const bool maskcol = (k0 + m) >= vlen;
#pragma unroll
        for (int r = 0; r < 8; ++r) {
            float sv = s[r] * SCALE;
            s[r] = maskcol ? NEGV : sv;
        }

        // ---- online softmax (rows live in 16-lane halves) ----
#pragma unroll
        for (int r = 0; r < 8; ++r) {
            float x = s[r];
            x = fmaxf(x, __shfl_xor(x, 1));
            x = fmaxf(x, __shfl_xor(x, 2));
            x = fmaxf(x, __shfl_xor(x, 4));
            x = fmaxf(x, __shfl_xor(x, 8));
            const float mnew = fmaxf(mrow[r], x);
            const float corr = __expf(mrow[r] - mnew);
            mrow[r] = mnew;
            float p = __expf(s[r] - mnew);
            s[r] = p;
            float ps = p;
            ps += __shfl_xor(ps, 1);
            ps += __shfl_xor(ps, 2);
            ps += __shfl_xor(ps, 4);
            ps += __shfl_xor(ps, 8);
            lrow[r] = lrow[r] * corr + ps;
#pragma unroll
            for (int t = 0; t < 8; ++t) acc[t][r] *= corr;
        }

        // ---- transpose P through per-wave LDS (C-layout -> A-layout) ----
#pragma unroll
        for (int r = 0; r < 8; ++r)
            pw[(r + hi * 8) * 18 + m] = s[r];
        asm volatile("s_wait_dscnt 0x0" ::: "memory");
        v2f pa[4];
#pragma unroll
        for (int kk = 0; kk < 4; ++kk)
            pa[kk] = *(const v2f*)(pw + m * 18 + 4 * kk + 2 * hi);
        asm volatile("" ::: "memory");

        // ---- O += P * V : 8 independent chains, V operands for d-tile t+1
        //      loaded while d-tile t's WMMAs execute ----
        const float* vt0 = vs + m;
        v2f vvA[4], vvB[4];
#pragma unroll
        for (int kk = 0; kk < 4; ++kk) {
            vvA[kk].x = vt0[(4 * kk + 2 * hi) * VSTR];
            vvA[kk].y = vt0[(4 * kk + 2 * hi + 1) * VSTR];
        }
#pragma unroll
        for (int t = 0; t < 8; ++t) {
            v2f* cur = (t & 1) ? vvB : vvA;
            v2f* nxt = (t & 1) ? vvA : vvB;
            if (t < 7) {
                const float* vtn = vs + (t + 1) * 16 + m;
#pragma unroll
                for (int kk = 0; kk < 4; ++kk) {
                    nxt[kk].x = vtn[(4 * kk + 2 * hi) * VSTR];
                    nxt[kk].y = vtn[(4 * kk + 2 * hi + 1) * VSTR];
                }
            }
#pragma unroll
            for (int kk = 0; kk < 4; ++kk)
                acc[t] = __builtin_amdgcn_wmma_f32_16x16x4_f32(
                             false, pa[kk], false, cur[kk], (short)0, acc[t],
                             false, false);
        }
        __syncthreads();   // all waves done reading before buffer is reused
    }

    // ---- normalize and store ----
#pragma unroll
    for (int r = 0; r < 8; ++r) {
        const float inv = 1.0f / lrow[r];
        const int   row = r + hi * 8;
#pragma unroll
        for (int t = 0; t < 8; ++t)
            Ob[(size_t)row * D_ + t * 16 + m] = acc[t][r] * inv;
    }
}

extern "C" void kernel_launch(void* const* d_in, const int* in_sizes, int n_in,
                              void* d_out, int out_size, void* d_ws, size_t ws_size,
                              hipStream_t stream) {
    (void)in_sizes; (void)n_in; (void)out_size; (void)d_ws; (void)ws_size;
    const float* q  = (const float*)d_in[0];
    const float* k  = (const float*)d_in[1];
    const float* v  = (const float*)d_in[2];
    const int*   vl = (const int*)d_in[3];
    float*       o  = (float*)d_out;

    const int waves  = (B_ * Q_) / 16; // 2048 waves, 16 query rows each
    const int blocks = waves / 8;      // 8 waves (one batch slice) per block
    attn_f32_wmma_tdm_kernel<<<blocks, 256, 0, stream>>>(q, k, v, vl, o);
}